// GRU_31439160607325
// MI455X (gfx1250) — compile-verified
//
#include <hip/hip_runtime.h>
#include <hip/hip_bf16.h>

// Problem constants (match reference)
#define Hd 100       // hidden size
#define Id 23        // input size
#define Td 512       // time steps
#define Bd 512       // batch
#define TILE_B 16    // batch rows per workgroup
#define KPAD 124     // 23 (x) + 1 pad + 100 (h)
#define KSTRIDE 132  // Wt row stride (pad 124 -> 132 words: bank-conflict-free b64 loads)
#define NCOLS 400    // r(100) z(100) gh_n(100) xp_n(100)
#define NTILES 25    // 400/16
#define ASTRIDE 128  // A_lds row stride

typedef __attribute__((ext_vector_type(2))) float v2f;
typedef __attribute__((ext_vector_type(8))) float v8f;

__device__ __forceinline__ float sigmoid_f(float v) {
    return 1.0f / (1.0f + __expf(-v));
}
__device__ __forceinline__ float tanh_f(float v) {
    // 1 - 2/(e^{2v}+1): saturates cleanly to +-1 without inf/inf NaN
    float e = __expf(2.0f * v);
    return 1.0f - 2.0f / (e + 1.0f);
}

// One workgroup: (direction, 16-batch-row tile). Runs the full T=512 scan
// with weights + state resident in LDS; gate pre-activations via fp32 WMMA.
__global__ void __launch_bounds__(256)
gru_scan_kernel(const float* __restrict__ x,
                const float* __restrict__ w_ih_f, const float* __restrict__ w_hh_f,
                const float* __restrict__ b_ih_f, const float* __restrict__ b_hh_f,
                const float* __restrict__ w_ih_b, const float* __restrict__ w_hh_b,
                const float* __restrict__ b_ih_b, const float* __restrict__ b_hh_b,
                float* __restrict__ hbuf)   // [2][Bd][Hd] final hidden states
{
    extern __shared__ float smem[];
    float* Wt_lds   = smem;                          // [NCOLS][KSTRIDE]  (transposed!)
    float* A_lds    = Wt_lds + NCOLS * KSTRIDE;      // [TILE_B][ASTRIDE]
    float* G_lds    = A_lds + TILE_B * ASTRIDE;      // [TILE_B][NCOLS]
    float* bias_lds = G_lds + TILE_B * NCOLS;        // [NCOLS]

    const int tid  = threadIdx.x;
    const int dir  = blockIdx.y;          // 0 = forward, 1 = backward
    const int b0   = blockIdx.x * TILE_B; // first batch row of this tile

    const float* w_ih = dir ? w_ih_b : w_ih_f;   // [3H][I]
    const float* w_hh = dir ? w_hh_b : w_hh_f;   // [3H][H]
    const float* b_ih = dir ? b_ih_b : b_ih_f;   // [3H]
    const float* b_hh = dir ? b_hh_b : b_hh_f;   // [3H]

    // ---- Stage combined weight matrix into LDS, TRANSPOSED: Wt[c][k] = W[k][c].
    // K rows 0..22 : input features (w_ih), row 23: zero pad, rows 24..123: hidden (w_hh),
    // rows 124..131: stride pad (zero).
    // N cols 0..99 r (ih+hh) | 100..199 z (ih+hh) | 200..299 gh_n (hh only)
    // N cols 300..399 xp_n (ih only)
    for (int idx = tid; idx < NCOLS * KSTRIDE; idx += 256) {
        int c = idx / KSTRIDE, k = idx % KSTRIDE;
        float w = 0.0f;
        if (k < Id) {                         // x rows, input feature i=k
            if (c < 200)        w = w_ih[c * Id + k];
            else if (c >= 300)  w = w_ih[(c - 100) * Id + k];  // n-gate rows 200..299
        } else if (k >= Id + 1 && k < KPAD) { // h rows, j = k-24
            int j = k - (Id + 1);
            if (c < 300)        w = w_hh[c * Hd + j];
        }
        Wt_lds[idx] = w;
    }
    for (int c = tid; c < NCOLS; c += 256) {
        float bv;
        if (c < 200)        bv = b_ih[c] + b_hh[c];   // r,z: biases sum
        else if (c < 300)   bv = b_hh[c];             // gh_n: hh bias only
        else                bv = b_ih[c - 100];       // xp_n: ih bias only
        bias_lds[c] = bv;
    }
    // h0 = 0 and zero the K padding column
    for (int idx = tid; idx < TILE_B * ASTRIDE; idx += 256) A_lds[idx] = 0.0f;
    __syncthreads();

    const int lane = tid & 31;
    const int wave = tid >> 5;
    const int mrow = lane & 15;      // A row / B-N lane / C column lane
    const int hi   = lane >> 4;      // half-wave selector per ISA layouts

    // Per-lane fragment base pointers: all K-loop accesses become small
    // immediate offsets -> ds_load_2addr_b64, no per-k address arithmetic.
    const float* aptr = &A_lds[mrow * ASTRIDE + 2 * hi];

    for (int t = 0; t < Td; ++t) {
        const int tt = dir ? (Td - 1 - t) : t;

        // ---- Load x_t tile into A_lds[:,0:23]
        for (int idx = tid; idx < TILE_B * Id; idx += 256) {
            int m = idx / Id, i = idx % Id;
            A_lds[m * ASTRIDE + i] = x[((size_t)(b0 + m) * Td + tt) * Id + i];
        }
        // gfx1250 prefetch of next step's x rows
        if (t + 1 < Td && tid < TILE_B) {
            int tn = dir ? (Td - 2 - t) : (t + 1);
            __builtin_prefetch(&x[((size_t)(b0 + tid) * Td + tn) * Id], 0, 3);
        }
        __syncthreads();

        // ---- Gate pre-activations: [16,124] x [124,400] via v_wmma_f32_16x16x4_f32
        for (int nt = wave; nt < NTILES; nt += 8) {
            const int N0 = nt * 16;
            const float* wptr = &Wt_lds[(N0 + mrow) * KSTRIDE + 2 * hi];
            v8f acc = {};
            #pragma unroll
            for (int k0 = 0; k0 < KPAD; k0 += 4) {
                v2f a = *(const v2f*)(aptr + k0);   // A[m][k0+2hi .. +1]
                v2f b = *(const v2f*)(wptr + k0);   // B[k0+2hi .. +1][n] (transposed)
                acc = __builtin_amdgcn_wmma_f32_16x16x4_f32(
                        false, a, false, b, (short)0, acc, false, false);
            }
            const int col = N0 + mrow;
            const float bb = bias_lds[col];
            #pragma unroll
            for (int v = 0; v < 8; ++v) {
                int row = v + 8 * hi;                                // C row = v + 8*hi
                G_lds[row * NCOLS + col] = acc[v] + bb;
            }
        }
        __syncthreads();

        // ---- GRU gate update (r,z,n), h lives in A_lds[:,24:124]
        for (int idx = tid; idx < TILE_B * Hd; idx += 256) {
            int m = idx / Hd, j = idx % Hd;
            const float* g = &G_lds[m * NCOLS];
            float r = sigmoid_f(g[j]);
            float z = sigmoid_f(g[100 + j]);
            float n = tanh_f(g[300 + j] + r * g[200 + j]);
            float hprev = A_lds[m * ASTRIDE + (Id + 1) + j];
            A_lds[m * ASTRIDE + (Id + 1) + j] = (1.0f - z) * n + z * hprev;
        }
        __syncthreads();
    }

    // ---- Emit final hidden state to workspace: hbuf[dir][b][j]
    for (int idx = tid; idx < TILE_B * Hd; idx += 256) {
        int m = idx / Hd, j = idx % Hd;
        hbuf[((size_t)dir * Bd + (b0 + m)) * Hd + j] = A_lds[m * ASTRIDE + (Id + 1) + j];
    }
}

// out[b][o] = [h_fw|h_bw] @ fc_w^T + fc_b   (tiny: 512x200x4)
__global__ void __launch_bounds__(256)
gru_fc_kernel(const float* __restrict__ hbuf,
              const float* __restrict__ fc_w,   // [4][200]
              const float* __restrict__ fc_b,   // [4]
              float* __restrict__ out)          // [Bd][4]
{
    int idx = blockIdx.x * blockDim.x + threadIdx.x;
    if (idx >= Bd * 4) return;
    int b = idx >> 2, o = idx & 3;
    const float* hf = hbuf + (size_t)b * Hd;
    const float* hb = hbuf + (size_t)(Bd + b) * Hd;
    const float* w  = fc_w + o * (2 * Hd);
    float acc = fc_b[o];
    #pragma unroll 4
    for (int j = 0; j < Hd; ++j) acc += hf[j] * w[j];
    #pragma unroll 4
    for (int j = 0; j < Hd; ++j) acc += hb[j] * w[Hd + j];
    out[(size_t)b * 4 + o] = acc;
}

extern "C" void kernel_launch(void* const* d_in, const int* in_sizes, int n_in,
                              void* d_out, int out_size, void* d_ws, size_t ws_size,
                              hipStream_t stream) {
    const float* x      = (const float*)d_in[0];
    const float* w_ih_f = (const float*)d_in[1];
    const float* w_hh_f = (const float*)d_in[2];
    const float* b_ih_f = (const float*)d_in[3];
    const float* b_hh_f = (const float*)d_in[4];
    const float* w_ih_b = (const float*)d_in[5];
    const float* w_hh_b = (const float*)d_in[6];
    const float* b_ih_b = (const float*)d_in[7];
    const float* b_hh_b = (const float*)d_in[8];
    const float* fc_w   = (const float*)d_in[9];
    const float* fc_b   = (const float*)d_in[10];

    float* hbuf = (float*)d_ws;   // 2*512*100 floats = 400 KB scratch

    const size_t lds_bytes =
        (size_t)(NCOLS * KSTRIDE + TILE_B * ASTRIDE + TILE_B * NCOLS + NCOLS) * sizeof(float);

    dim3 grid(Bd / TILE_B, 2);   // 32 batch tiles x 2 directions
    gru_scan_kernel<<<grid, 256, lds_bytes, stream>>>(
        x, w_ih_f, w_hh_f, b_ih_f, b_hh_f,
        w_ih_b, w_hh_b, b_ih_b, b_hh_b, hbuf);

    gru_fc_kernel<<<(Bd * 4 + 255) / 256, 256, 0, stream>>>(hbuf, fc_w, fc_b, (float*)d_out);
}